// TripletLoss_58231166599209
// MI455X (gfx1250) — compile-verified
//
#include <hip/hip_runtime.h>

typedef __attribute__((ext_vector_type(2))) float v2f;
typedef __attribute__((ext_vector_type(8))) float v8f;

#define DIM 512
#define TRIP_PER_WAVE 16
#define WAVES_PER_BLOCK 8
#define THREADS (WAVES_PER_BLOCK * 32)
#define TRIP_PER_BLOCK (WAVES_PER_BLOCK * TRIP_PER_WAVE)
#define MARGIN 1.0f

// Each wave32 owns 16 triplets. Per K-step of 4 columns it loads one float2
// per row (a, p, n), forms u = a-p and w = a-n, and issues two
// V_WMMA_F32_16X16X4_F32 Gram-matrix accumulations. For the f32 16x16x4
// shape the A(16x4) and B(4x16) VGPR images of U and U^T coincide lane-for-
// lane, so the same v2f feeds both operands. After 128 steps the accumulator
// diagonal holds d_ap / d_an for the 16 triplets.
__global__ __launch_bounds__(THREADS)
void TripletLoss_wmma_kernel(const float* __restrict__ batch,
                             const int* __restrict__ trip,
                             float* __restrict__ partial,
                             int nTrip) {
    __shared__ float wsum[WAVES_PER_BLOCK];
    const int lane = threadIdx.x & 31;
    const int wave = threadIdx.x >> 5;
    const int base = blockIdx.x * TRIP_PER_BLOCK + wave * TRIP_PER_WAVE;

    float loss = 0.0f;
    if (base < nTrip) {                      // wave-uniform branch
        const int m = lane & 15;             // row of the 16-triplet group
        int r = base + m;
        if (r >= nTrip) r = nTrip - 1;       // clamp (keeps EXEC full)
        const int i0 = trip[3 * r + 0];
        const int i1 = trip[3 * r + 1];
        const int i2 = trip[3 * r + 2];
        const float* __restrict__ A = batch + i0 * DIM;
        const float* __restrict__ P = batch + i1 * DIM;
        const float* __restrict__ N = batch + i2 * DIM;
        // Warm the back half of each gathered row (global_prefetch_b8).
        __builtin_prefetch(A + 256, 0, 3);
        __builtin_prefetch(P + 256, 0, 3);
        __builtin_prefetch(N + 256, 0, 3);

        // lanes 0-15 supply K = k0+0,k0+1 ; lanes 16-31 supply K = k0+2,k0+3
        const int koff = (lane >> 4) << 1;

        v8f cap = {};   // accumulates U * U^T  (diag = d_ap)
        v8f can = {};   // accumulates W * W^T  (diag = d_an)
#pragma unroll 8
        for (int k0 = 0; k0 < DIM; k0 += 4) {
            v2f a2 = *(const v2f*)(A + k0 + koff);
            v2f p2 = *(const v2f*)(P + k0 + koff);
            v2f n2 = *(const v2f*)(N + k0 + koff);
            v2f u = a2 - p2;
            v2f w = a2 - n2;
            cap = __builtin_amdgcn_wmma_f32_16x16x4_f32(
                false, u, false, u, (short)0, cap, false, false);
            can = __builtin_amdgcn_wmma_f32_16x16x4_f32(
                false, w, false, w, (short)0, can, false, false);
        }

        // Diagonal extraction from the 16x16 f32 C layout:
        //   triplet M (0..7)  lives in VGPR M  at lane M
        //   triplet M (8..15) lives in VGPR M-8 at lane M+16
        const bool active = (lane < 8) || (lane >= 24);
        const int sel = (lane < 8) ? lane : (lane - 24);
        float dap = 0.0f, dan = 0.0f;
#pragma unroll
        for (int j = 0; j < 8; ++j) {
            if (sel == j) { dap = cap[j]; dan = can[j]; }
        }
        const int mOut = (lane < 8) ? lane : (lane - 16);
        if (active && (base + mOut) < nTrip)
            loss = fmaxf(dap - dan + MARGIN, 0.0f);
    }

    // wave32 butterfly reduction
#pragma unroll
    for (int off = 16; off > 0; off >>= 1)
        loss += __shfl_xor(loss, off, 32);
    if (lane == 0) wsum[wave] = loss;
    __syncthreads();
    if (threadIdx.x == 0) {
        float s = 0.0f;
#pragma unroll
        for (int j = 0; j < WAVES_PER_BLOCK; ++j) s += wsum[j];
        partial[blockIdx.x] = s;
    }
}

// Deterministic fixed-order tree over the per-block partials.
__global__ __launch_bounds__(256)
void TripletLoss_reduce_kernel(const float* __restrict__ part,
                               float* __restrict__ out, int n) {
    __shared__ float s[256];
    const int t = threadIdx.x;
    s[t] = (t < n) ? part[t] : 0.0f;
    __syncthreads();
#pragma unroll
    for (int off = 128; off > 0; off >>= 1) {
        if (t < off) s[t] += s[t + off];
        __syncthreads();
    }
    if (t == 0) out[0] = s[0];
}

extern "C" void kernel_launch(void* const* d_in, const int* in_sizes, int n_in,
                              void* d_out, int out_size, void* d_ws, size_t ws_size,
                              hipStream_t stream) {
    const float* batch = (const float*)d_in[0];
    const int* trip    = (const int*)d_in[1];
    float* out         = (float*)d_out;
    float* partial     = (float*)d_ws;

    const int nTrip  = in_sizes[1] / 3;                       // 16384
    const int blocks = (nTrip + TRIP_PER_BLOCK - 1) / TRIP_PER_BLOCK;  // 128

    TripletLoss_wmma_kernel<<<blocks, THREADS, 0, stream>>>(batch, trip,
                                                            partial, nTrip);
    TripletLoss_reduce_kernel<<<1, 256, 0, stream>>>(partial, out, blocks);
}